// UniMPLayer_44813688766818
// MI455X (gfx1250) — compile-verified
//
#include <hip/hip_runtime.h>
#include <hip/hip_bf16.h>

// ---------------------------------------------------------------------------
// UniMP / TransformerConv layer for MI455X (gfx1250, wave32)
//   proj (WMMA f32 16x16x4) -> edge logits + seg-max -> exp/scatter -> finalize
// ---------------------------------------------------------------------------

#define HEADS   4
#define CH      16          // per-head channels
#define HC      64          // HEADS*CH
#define LN_EPS  1e-5f

typedef float v2f __attribute__((ext_vector_type(2)));
typedef float v8f __attribute__((ext_vector_type(8)));

// ---- ordered-uint encoding for float atomic max --------------------------
__device__ __forceinline__ unsigned f2ord(float f) {
    unsigned u = __float_as_uint(f);
    return (u & 0x80000000u) ? ~u : (u | 0x80000000u);
}
__device__ __forceinline__ float ord2f(unsigned u) {
    return (u & 0x80000000u) ? __uint_as_float(u & 0x7FFFFFFFu)
                             : __uint_as_float(~u);
}
#define ORD_NEG_INF 0x007FFFFFu   // f2ord(-inf)

// ---------------------------------------------------------------------------
// Kernel 0: init scratch (seg-max identity, zero sums / accumulators)
// ---------------------------------------------------------------------------
__global__ void unimp_init(unsigned* __restrict__ m_ord,
                           float* __restrict__ ssum,
                           float* __restrict__ acc, int N) {
    int t = blockIdx.x * blockDim.x + threadIdx.x;
    if (t < N * HEADS) { m_ord[t] = ORD_NEG_INF; ssum[t] = 0.f; }
    if (t < N * HC)    { acc[t] = 0.f; }
}

// ---------------------------------------------------------------------------
// Kernel 1: fused projections q/k/v/x_r via V_WMMA_F32_16X16X4_F32.
// One 16-node tile per block; wave w (of 4) computes projection w.
// C[16x16] tile per N-subtile, K=16 done as 4 chained K=4 WMMAs (full f32).
// Tile-full check is uniform -> scalar branch, stores use immediate offsets.
// ---------------------------------------------------------------------------
__global__ __launch_bounds__(128)
void unimp_proj(const float* __restrict__ x,
                const float* __restrict__ Wq, const float* __restrict__ bq,
                const float* __restrict__ Wk, const float* __restrict__ bk,
                const float* __restrict__ Wv, const float* __restrict__ bv,
                const float* __restrict__ Ws, const float* __restrict__ bs,
                float* __restrict__ q, float* __restrict__ k,
                float* __restrict__ v, float* __restrict__ xr, int N) {
    const int wave = threadIdx.x >> 5;        // 0..3 -> which projection
    const int lane = threadIdx.x & 31;
    const int node0 = blockIdx.x * 16;

    const float* W; const float* b; float* out;
    if      (wave == 0) { W = Wq; b = bq; out = q;  }
    else if (wave == 1) { W = Wk; b = bk; out = k;  }
    else if (wave == 2) { W = Wv; b = bv; out = v;  }
    else                { W = Ws; b = bs; out = xr; }

    const int mrow  = lane & 15;              // M (or N) index within tile
    const int khalf = lane >> 4;              // selects K pair {0,1} vs {2,3}
    const bool fullTile = (node0 + 16 <= N);  // uniform across wave

    // A fragments: 16x4 f32, lane L: M=L%16, VGPR0=K(base), VGPR1=K(base+1),
    // base = 4*kc + 2*(L/16).  (ISA 7.12.2, 32-bit A 16x4)
    int arow = node0 + mrow; if (arow >= N) arow = N - 1;   // clamped read
    const float* xrow = x + (size_t)arow * CH;
    v2f A[4];
#pragma unroll
    for (int kc = 0; kc < 4; ++kc) {
        const int k0 = kc * 4 + khalf * 2;
        A[kc].x = xrow[k0];
        A[kc].y = xrow[k0 + 1];
    }

#pragma unroll
    for (int nt = 0; nt < 4; ++nt) {          // four 16-col output subtiles
        v8f C = {};
#pragma unroll
        for (int kc = 0; kc < 4; ++kc) {
            const int k0 = kc * 4 + khalf * 2;
            v2f B;                             // B 4x16: lane L: N=L%16
            B.x = W[(size_t)k0       * HC + nt * 16 + mrow];
            B.y = W[(size_t)(k0 + 1) * HC + nt * 16 + mrow];
            C = __builtin_amdgcn_wmma_f32_16x16x4_f32(
                    false, A[kc], false, B, (short)0, C, false, false);
        }
        const float bias = b[nt * 16 + mrow];
        // C/D layout: VGPR r -> M = r + 8*(lane/16), N = lane%16
        // Common path: whole 16-row tile valid -> straight-line stores from
        // one base with immediate offsets (r*HC floats apart).
        float* obase = out + (size_t)(node0 + khalf * 8) * HC + nt * 16 + mrow;
        if (fullTile) {
#pragma unroll
            for (int r = 0; r < 8; ++r)
                obase[(size_t)r * HC] = C[r] + bias;
        } else {
#pragma unroll
            for (int r = 0; r < 8; ++r) {
                const int node = node0 + khalf * 8 + r;
                if (node < N)
                    obase[(size_t)r * HC] = C[r] + bias;
            }
        }
    }
}

// ---------------------------------------------------------------------------
// Kernel 2: per-(edge,head) attention logit + segment max (ordered-uint).
// ---------------------------------------------------------------------------
__global__ __launch_bounds__(256)
void unimp_alpha(const long long* __restrict__ ei,
                 const float* __restrict__ q, const float* __restrict__ k,
                 float* __restrict__ alpha, unsigned* __restrict__ m_ord,
                 int E) {
    int t = blockIdx.x * blockDim.x + threadIdx.x;
    if (t >= E * HEADS) return;
    const int e = t >> 2, h = t & 3;
    const int s = (int)ei[e];
    const int d = (int)ei[E + e];
    const float4* qr = (const float4*)(q + (size_t)d * HC + h * CH);
    const float4* kr = (const float4*)(k + (size_t)s * HC + h * CH);
    float a = 0.f;
#pragma unroll
    for (int i = 0; i < 4; ++i) {
        const float4 qa = qr[i], kb = kr[i];
        a += qa.x * kb.x + qa.y * kb.y + qa.z * kb.z + qa.w * kb.w;
    }
    a *= 0.25f;                               // 1/sqrt(16)
    alpha[t] = a;
    atomicMax(&m_ord[d * HEADS + h], f2ord(a));
}

// ---------------------------------------------------------------------------
// Kernel 3: per-(edge,head) exp weight, scatter denominator + numerator.
// (normalization folded into finalize: out = acc / (s + 1e-16))
// ---------------------------------------------------------------------------
__global__ __launch_bounds__(256)
void unimp_scatter(const long long* __restrict__ ei,
                   const float* __restrict__ v, const float* __restrict__ alpha,
                   const unsigned* __restrict__ m_ord,
                   float* __restrict__ ssum, float* __restrict__ acc, int E) {
    int t = blockIdx.x * blockDim.x + threadIdx.x;
    if (t >= E * HEADS) return;
    const int e = t >> 2, h = t & 3;
    const int s = (int)ei[e];
    const int d = (int)ei[E + e];
    const float m = ord2f(m_ord[d * HEADS + h]);
    const float w = __expf(alpha[t] - m);
    atomicAdd(&ssum[d * HEADS + h], w);
    const float4* vr = (const float4*)(v + (size_t)s * HC + h * CH);
    float*        ar = acc + (size_t)d * HC + h * CH;
#pragma unroll
    for (int i = 0; i < 4; ++i) {
        const float4 vv = vr[i];
        atomicAdd(&ar[4 * i + 0], w * vv.x);
        atomicAdd(&ar[4 * i + 1], w * vv.y);
        atomicAdd(&ar[4 * i + 2], w * vv.z);
        atomicAdd(&ar[4 * i + 3], w * vv.w);
    }
}

// ---------------------------------------------------------------------------
// Kernel 4: per-node finalize. One wave32 per node, 2 channels per lane.
// beta gate (Wbeta[192,1]) + LayerNorm + ReLU via shfl_xor wave reductions.
// ---------------------------------------------------------------------------
__global__ __launch_bounds__(256)
void unimp_finalize(const float* __restrict__ acc, const float* __restrict__ ssum,
                    const float* __restrict__ xr, const float* __restrict__ Wbeta,
                    const float* __restrict__ ln_g, const float* __restrict__ ln_b,
                    float* __restrict__ out, int N) {
    const int wave = threadIdx.x >> 5;
    const int lane = threadIdx.x & 31;
    const int node = blockIdx.x * 8 + wave;
    if (node >= N) return;

    float o[2], r[2];
#pragma unroll
    for (int i = 0; i < 2; ++i) {
        const int c = lane + 32 * i;
        const int h = c >> 4;
        const float s = ssum[node * HEADS + h];
        o[i] = acc[(size_t)node * HC + c] / (s + 1e-16f);
        r[i] = xr[(size_t)node * HC + c];
    }
    // beta gate: sigmoid( [out | x_r | out-x_r] . Wbeta )
    float dotv = 0.f;
#pragma unroll
    for (int i = 0; i < 2; ++i) {
        const int c = lane + 32 * i;
        dotv += o[i] * Wbeta[c] + r[i] * Wbeta[HC + c]
              + (o[i] - r[i]) * Wbeta[2 * HC + c];
    }
#pragma unroll
    for (int off = 16; off > 0; off >>= 1) dotv += __shfl_xor(dotv, off, 32);
    const float g = 1.f / (1.f + __expf(-dotv));
#pragma unroll
    for (int i = 0; i < 2; ++i) o[i] = g * r[i] + (1.f - g) * o[i];

    // LayerNorm over 64 channels (population variance) + ReLU
    float sum = o[0] + o[1];
    float sq  = o[0] * o[0] + o[1] * o[1];
#pragma unroll
    for (int off = 16; off > 0; off >>= 1) {
        sum += __shfl_xor(sum, off, 32);
        sq  += __shfl_xor(sq,  off, 32);
    }
    const float mu  = sum * (1.f / HC);
    const float var = sq * (1.f / HC) - mu * mu;
    const float inv = rsqrtf(var + LN_EPS);
#pragma unroll
    for (int i = 0; i < 2; ++i) {
        const int c = lane + 32 * i;
        const float y = (o[i] - mu) * inv * ln_g[c] + ln_b[c];
        out[(size_t)node * HC + c] = fmaxf(y, 0.f);
    }
}

// ---------------------------------------------------------------------------
extern "C" void kernel_launch(void* const* d_in, const int* in_sizes, int n_in,
                              void* d_out, int out_size, void* d_ws, size_t ws_size,
                              hipStream_t stream) {
    const float*     x    = (const float*)d_in[0];
    const long long* ei   = (const long long*)d_in[1];   // int64 [2,E]
    const float* Wq = (const float*)d_in[2],  *bq = (const float*)d_in[3];
    const float* Wk = (const float*)d_in[4],  *bk = (const float*)d_in[5];
    const float* Wv = (const float*)d_in[6],  *bv = (const float*)d_in[7];
    const float* Ws = (const float*)d_in[8],  *bs = (const float*)d_in[9];
    const float* Wbeta = (const float*)d_in[10];
    const float* lng = (const float*)d_in[11], *lnb = (const float*)d_in[12];

    const int N = in_sizes[0] / CH;       // 100000
    const int E = in_sizes[1] / 2;        // 1600000

    // workspace layout (floats): q,k,v,xr,acc [N*64 each]; m_ord,s [N*4]; alpha [E*4]
    float* q   = (float*)d_ws;
    float* k   = q   + (size_t)N * HC;
    float* v   = k   + (size_t)N * HC;
    float* xr  = v   + (size_t)N * HC;
    float* acc = xr  + (size_t)N * HC;
    unsigned* m_ord = (unsigned*)(acc + (size_t)N * HC);
    float* ssum  = (float*)(m_ord + (size_t)N * HEADS);
    float* alpha = ssum + (size_t)N * HEADS;

    const int initN = N * HC;
    unimp_init<<<(initN + 255) / 256, 256, 0, stream>>>(m_ord, ssum, acc, N);

    unimp_proj<<<(N + 15) / 16, 128, 0, stream>>>(
        x, Wq, bq, Wk, bk, Wv, bv, Ws, bs, q, k, v, xr, N);

    const int EH = E * HEADS;
    unimp_alpha<<<(EH + 255) / 256, 256, 0, stream>>>(ei, q, k, alpha, m_ord, E);
    unimp_scatter<<<(EH + 255) / 256, 256, 0, stream>>>(ei, v, alpha, m_ord, ssum, acc, E);
    unimp_finalize<<<(N + 7) / 8, 256, 0, stream>>>(
        acc, ssum, xr, Wbeta, lng, lnb, (float*)d_out, N);
}